// fd2d_48747878810210
// MI455X (gfx1250) — compile-verified
//
// MI455X / gfx1250 persistent single-WGP elastic FDTD, v3.
// Latency-bound (1500 serial steps, 106x200 grid) -> ONE 1024-thread workgroup
// (32 x wave32 on one WGP), hardware s_barrier for the 3000 phase syncs.
// LDS (320 KB/WGP) holds the HOT read-write fields vx/vz/txz (254 KB) +
// source wavelet; read-only coefficients live in global ws (L0-cached).
// CDNA5 paths: tensor_load_to_lds (TDM, TENSORcnt) for initial field staging,
// global_load_async_to_lds_b32 (ASYNCcnt) for the wavelet, s_wait_tensorcnt /
// s_wait_asynccnt, global_prefetch_b8.
#include <hip/hip_runtime.h>

#define NPAD   40
#define NZP    106
#define NXP    200
#define NCELL  (NZP * NXP)   // 21200
#define NXINT  120
#define RECROW (NPAD + 1)
#define SMAX   1504
#define NTHR   1024

#if __has_builtin(__builtin_amdgcn_tensor_load_to_lds)
#define HAVE_TDM 1
typedef unsigned int v4u __attribute__((ext_vector_type(4)));
typedef int          v8i __attribute__((ext_vector_type(8)));
typedef int          v4i __attribute__((ext_vector_type(4)));
#else
#define HAVE_TDM 0
#endif

__device__ __forceinline__ float damp1d(int i, int n) {
    float l = 1.0f, r = 1.0f;
    if (i < NPAD) {
        float a = 0.015f * (float)(NPAD - i);
        l = expf(-a * a);
    }
    if (i >= n - NPAD) {
        float a = 0.015f * (float)(i - (n - NPAD - 1));
        r = expf(-a * a);
    }
    return l * r;
}

#if HAVE_TDM
// TDM descriptor per CDNA5 ISA: 2-D tile (200 x 106 f32,
// contiguous) from global to LDS. Issued once per wave; tracked on TENSORcnt.
// This toolchain exposes the 6-arg builtin: (g0, g1, g2, g3, g_extra, cpol).
__device__ __forceinline__ void tdm_load_plane(unsigned lds_off, const float* gptr) {
    unsigned long long ga = (unsigned long long)gptr;
    v4u g0;
    g0.x = 1u;                                          // count=1, user D#
    g0.y = lds_off;                                     // lds_addr (bytes)
    g0.z = (unsigned)(ga & 0xFFFFFFFFu);                // global_addr[31:0]
    g0.w = (unsigned)((ga >> 32) & 0x1FFFFFFu)          // global_addr[56:32]
         | (2u << 30);                                  // type=2 ("image")
    v8i g1;
    g1[0] = 0x20000;                                    // data_size=4B; no mask/pad
    g1[1] = (int)(((unsigned)NXP & 0xFFFFu) << 16);     // tensor_dim0 lo16 @ [63:48]
    g1[2] = (int)(((unsigned)NZP & 0xFFFFu) << 16);     // dim0 hi=0 | tensor_dim1 lo16
    g1[3] = (int)(((unsigned)NXP) << 16);               // dim1 hi=0 | tile_dim0=200
    g1[4] = NZP;                                        // tile_dim1=106, tile_dim2=0
    g1[5] = NXP;                                        // tensor_dim0_stride=200
    g1[6] = 0;
    g1[7] = 0;
    v4i z4 = {0, 0, 0, 0};
    v8i z8 = {0, 0, 0, 0, 0, 0, 0, 0};
    __builtin_amdgcn_tensor_load_to_lds(g0, g1, z4, z4, z8, 0);
}
#endif

extern "C" __global__ __launch_bounds__(NTHR, 1)
void fd2d_elastic_persistent(const float* __restrict__ vp,
                             const float* __restrict__ vs,
                             const float* __restrict__ den,
                             const float* __restrict__ ivec,
                             const float* __restrict__ temp,
                             const float* __restrict__ s,
                             const int*   __restrict__ nt_p,
                             const int*   __restrict__ isz_p,
                             const int*   __restrict__ isx_p,
                             float*       __restrict__ out,
                             float*       __restrict__ ws)
{
    extern __shared__ float lds[];
    float* s_vx  = lds;                 // NCELL  (hot: 5 touches/step)
    float* s_vz  = lds + NCELL;         // NCELL  (hot: 5 touches/step + record)
    float* s_txz = lds + 2 * NCELL;     // NCELL  (hot: 5 touches/step)
    float* s_src = lds + 3 * NCELL;     // SMAX

    const int tid = threadIdx.x;
    const int nt  = nt_p[0];
    const int isz = isz_p[0];
    const int isx = isx_p[0];
    const int src_idx = isz * NXP + isx;
    const float dt  = ivec[2];
    const float dtx = ivec[3];

    // txx/tzz accumulate in-place in the output buffer (uu1 planes 2,3).
    float* out_vx  = out;
    float* out_vz  = out + 1 * NCELL;
    float* txx     = out + 2 * NCELL;
    float* tzz     = out + 3 * NCELL;
    float* out_txz = out + 4 * NCELL;
    float* seismo  = out + 5 * NCELL;
    float* illum   = out + 5 * NCELL + (size_t)nt * NXINT;
    // Read-only coefficients -> workspace (L0-cached stream, never written).
    float* ca   = ws;
    float* cl   = ws + 1 * NCELL;
    float* cm   = ws + 2 * NCELL;
    float* bb   = ws + 3 * NCELL;
    float* damp = ws + 4 * NCELL;

    __builtin_prefetch(den + tid, 0, 0);
    __builtin_prefetch(temp + tid, 0, 0);

#if HAVE_TDM
    // Wave 0 DMAs the initial vx/vz/txz planes (temp[0],temp[1],temp[4]) to LDS.
    if (tid < 32) {
        tdm_load_plane((unsigned)(size_t)s_vx,  temp);
        tdm_load_plane((unsigned)(size_t)s_vz,  temp + 1 * NCELL);
        tdm_load_plane((unsigned)(size_t)s_txz, temp + 4 * NCELL);
    }
#endif

    // Stage the source wavelet via CDNA5 async global->LDS copies (ASYNCcnt),
    // overlapping with the coefficient ALU work below.
    for (int i = tid; i < nt; i += NTHR) {
        unsigned           ldsoff = (unsigned)(size_t)(&s_src[i]);
        unsigned long long gaddr  = (unsigned long long)(s + i);
        asm volatile("global_load_async_to_lds_b32 %0, %1, off"
                     :: "v"(ldsoff), "v"(gaddr) : "memory");
    }

    // Coefficient setup + txx/tzz/illum init (overlaps TDM + async copies).
    for (int i = tid; i < NCELL; i += NTHR) {
        int z = i / NXP;
        int x = i - z * NXP;
        int zi = z - NPAD; zi = zi < 0 ? 0 : (zi > 25  ? 25  : zi);
        int xi = x - NPAD; xi = xi < 0 ? 0 : (xi > 119 ? 119 : xi);
        float d   = den[i];
        float vpe = vp[zi * 120 + xi];
        float vse = vs[zi * 120 + xi];
        float cav = vpe * vpe * d;
        float cmv = vse * vse * d;
        ca[i]   = cav;
        cl[i]   = cav - 2.0f * cmv;
        cm[i]   = cmv;
        bb[i]   = dtx / d;
        damp[i] = damp1d(z, NZP) * damp1d(x, NXP);
        txx[i]  = temp[2 * NCELL + i];
        tzz[i]  = temp[3 * NCELL + i];
        illum[i] = 0.0f;
#if !HAVE_TDM
        s_vx[i]  = temp[i];
        s_vz[i]  = temp[NCELL + i];
        s_txz[i] = temp[4 * NCELL + i];
#endif
    }

    // Drain the CDNA5 copy engines, then publish workgroup-wide.
    asm volatile("s_wait_asynccnt 0" ::: "memory");
#if HAVE_TDM
    if (tid < 32) __builtin_amdgcn_s_wait_tensorcnt(0);
#endif
    __syncthreads();

    // --- Time loop: 2 hardware barriers per step on a single WGP.
    for (int t = 0; t < nt; ++t) {
        const float src = s_src[t];

        // Phase 1: velocity update (reads stresses of step t-1).
        for (int i = tid; i < NCELL; i += NTHR) {
            int z = i / NXP;
            int x = i - z * NXP;
            int ixp = (x == NXP - 1) ? (i - (NXP - 1)) : (i + 1);        // roll -1 in x
            int ixm = (x == 0)       ? (i + (NXP - 1)) : (i - 1);        // roll +1 in x
            int izp = (z == NZP - 1) ? (i - (NZP - 1) * NXP) : (i + NXP);
            int izm = (z == 0)       ? (i + (NZP - 1) * NXP) : (i - NXP);
            float bv   = bb[i];
            float dv   = damp[i];
            float txzi = s_txz[i];
            float nvx = (s_vx[i] + bv * ((txx[ixp] - txx[i]) + (txzi - s_txz[izm]))) * dv;
            float nvz = (s_vz[i] + bv * ((tzz[izp] - tzz[i]) + (txzi - s_txz[ixm]))) * dv;
            s_vx[i] = nvx;
            s_vz[i] = nvz;
        }
        __syncthreads();

        // Record receiver row (post-update vz, straight from LDS).
        if (tid < NXINT)
            seismo[(size_t)t * NXINT + tid] = s_vz[RECROW * NXP + NPAD + tid];

        // Phase 2: stress update + source injection + illumination.
        for (int i = tid; i < NCELL; i += NTHR) {
            int z = i / NXP;
            int x = i - z * NXP;
            int ixp = (x == NXP - 1) ? (i - (NXP - 1)) : (i + 1);
            int ixm = (x == 0)       ? (i + (NXP - 1)) : (i - 1);
            int izp = (z == NZP - 1) ? (i - (NZP - 1) * NXP) : (i + NXP);
            int izm = (z == 0)       ? (i + (NZP - 1) * NXP) : (i - NXP);
            float vxi = s_vx[i], vzi = s_vz[i];
            float exx = vxi - s_vx[ixm];
            float ezz = vzi - s_vz[izm];
            float divv = exx + ezz;
            float cav = ca[i], clv = cl[i], cmv = cm[i], dv = damp[i];
            float ntxx = (txx[i] + dtx * (cav * exx + clv * ezz)) * dv;
            float ntzz = (tzz[i] + dtx * (clv * exx + cav * ezz)) * dv;
            float ntxz = (s_txz[i] + dtx * cmv * ((s_vx[izp] - vxi) + (s_vz[ixp] - vzi))) * dv;
            if (i == src_idx) {            // source injected by the owning thread
                ntxx += src * dt;
                ntzz += src * dt;
            }
            txx[i]   = ntxx;
            tzz[i]   = ntzz;
            s_txz[i] = ntxz;
            illum[i] += divv * divv;
        }
        __syncthreads();
    }

    // Final: spill the LDS-resident fields to the output (uu1 planes 0,1,4).
    for (int i = tid; i < NCELL; i += NTHR) {
        out_vx[i]  = s_vx[i];
        out_vz[i]  = s_vz[i];
        out_txz[i] = s_txz[i];
    }
}

extern "C" void kernel_launch(void* const* d_in, const int* in_sizes, int n_in,
                              void* d_out, int out_size, void* d_ws, size_t ws_size,
                              hipStream_t stream) {
    (void)in_sizes; (void)n_in; (void)out_size; (void)ws_size;
    const size_t shmem = (size_t)(3 * NCELL + SMAX) * sizeof(float); // ~260 KB of 320 KB WGP LDS
    fd2d_elastic_persistent<<<dim3(1), dim3(NTHR), shmem, stream>>>(
        (const float*)d_in[0],  // vp
        (const float*)d_in[1],  // vs
        (const float*)d_in[2],  // den
        (const float*)d_in[3],  // input_vector
        (const float*)d_in[4],  // temp
        (const float*)d_in[5],  // s
        (const int*)d_in[6],    // nt
        (const int*)d_in[7],    // isz
        (const int*)d_in[8],    // isx
        (float*)d_out,
        (float*)d_ws);
}